// PoseEstimator_377957122581
// MI455X (gfx1250) — compile-verified
//
#include <hip/hip_runtime.h>

// CDNA5 / gfx1250, wave32. V_WMMA_F32_16X16X4_F32 (K=4 == homogeneous 4-vec)
// computes 16x16 tiles of pairwise dot products, fused with a squared-distance
// min-reduction (sqrt hoisted outside the min: sqrt is monotonic, so
// min(sqrt(x)) == sqrt(min(x))). The 8192x8192 distance matrix is never
// materialized; all point data is precomputed as float4 (x,y,z,|p|^2+1) and
// stays L2-resident.

typedef __attribute__((ext_vector_type(2))) float v2f;
typedef __attribute__((ext_vector_type(8))) float v8f;

#define NPTS 8192
#define TILE 16
#define NBLK (NPTS / TILE)   // 512 row strips per directed pass

// ws layout (floats):
//   [0..15]    T_base   (row-major 4x4)
//   [16..31]   T_child  = rel @ T_base
//   [32..47]   T_identity
//   [48..2095] partial sums: 4 directed passes x NBLK
//   [2112.. )  4 transformed point clouds as float4: cad0', cam0', cad1', cam1'
#define WS_PART   48
#define WS_PTS    2112                  // float offset, 16B aligned
#define WS_CLOUD  (NPTS * 4)            // floats per cloud

// ---------------------------------------------------------------------------
// Transform prep: quat->rot, revolute screw transform, 4x4 matmul. Scalar.
// ---------------------------------------------------------------------------
__global__ void pe_prep(const float* __restrict__ xyz, const float* __restrict__ rpy,
                        const float* __restrict__ quat, const float* __restrict__ base_t,
                        const float* __restrict__ joint_state,
                        float* __restrict__ ws, float* __restrict__ out)
{
    if (threadIdx.x != 0 || blockIdx.x != 0) return;

    float a = quat[0], b = quat[1], c = quat[2], d = quat[3];
    float inv = rsqrtf(a * a + b * b + c * c + d * d);
    a *= inv; b *= inv; c *= inv; d *= inv;
    float R[9] = {
        1.f - 2.f*c*c - 2.f*d*d, 2.f*b*c - 2.f*a*d,       2.f*a*c + 2.f*b*d,
        2.f*b*c + 2.f*a*d,       1.f - 2.f*b*b - 2.f*d*d, 2.f*c*d - 2.f*a*b,
        2.f*b*d - 2.f*a*c,       2.f*a*b + 2.f*c*d,       1.f - 2.f*b*b - 2.f*c*c };

    float Tb[16];
    for (int i = 0; i < 3; i++) {
        for (int j = 0; j < 3; j++) Tb[i * 4 + j] = R[i * 3 + j];
        Tb[i * 4 + 3] = base_t[i];
    }
    Tb[12] = 0.f; Tb[13] = 0.f; Tb[14] = 0.f; Tb[15] = 1.f;

    // revolute screw transform (anchor = xyz, axis = rpy, unnormalized: matches ref)
    float ax = xyz[0], ay = xyz[1], az = xyz[2];
    float u = rpy[0], v = rpy[1], w = rpy[2];
    float th = joint_state[0];
    float cs = cosf(th), sn = sinf(th), oc = 1.f - cs;
    float Tr[16];
    Tr[0]  = u*u + (v*v + w*w)*cs;  Tr[1]  = u*v*oc - w*sn;  Tr[2]  = u*w*oc + v*sn;
    Tr[3]  = (ax*(v*v + w*w) - u*(ay*v + az*w))*oc + (ay*w - az*v)*sn;
    Tr[4]  = u*v*oc + w*sn;  Tr[5]  = v*v + (u*u + w*w)*cs;  Tr[6]  = v*w*oc - u*sn;
    Tr[7]  = (ay*(u*u + w*w) - v*(ax*u + az*w))*oc + (az*u - ax*w)*sn;
    Tr[8]  = u*w*oc - v*sn;  Tr[9]  = v*w*oc + u*sn;  Tr[10] = w*w + (u*u + v*v)*cs;
    Tr[11] = (az*(u*u + v*v) - w*(ax*u + ay*v))*oc + (ax*v - ay*u)*sn;
    Tr[12] = 0.f; Tr[13] = 0.f; Tr[14] = 0.f; Tr[15] = 1.f;

    float Tc[16];
    for (int i = 0; i < 4; i++)
        for (int j = 0; j < 4; j++) {
            float s = 0.f;
            for (int k = 0; k < 4; k++) s += Tr[i * 4 + k] * Tb[k * 4 + j];
            Tc[i * 4 + j] = s;
        }

    for (int i = 0; i < 16; i++) {
        ws[i]      = Tb[i];
        ws[16 + i] = Tc[i];
        ws[32 + i] = (i % 5 == 0) ? 1.f : 0.f;   // identity
        out[3 + i]  = Tb[i];                     // base_transform
        out[19 + i] = Tr[i];                     // relative_transform_all[0]
    }
}

// ---------------------------------------------------------------------------
// Apply 3x4 transform to a cloud, store (x', y', z', |p'|^2 + 1) as float4.
// ---------------------------------------------------------------------------
__global__ __launch_bounds__(256)
void pe_xform(const float* __restrict__ pts, const float* __restrict__ T,
              float4* __restrict__ dst)
{
    int i = blockIdx.x * blockDim.x + threadIdx.x;
    if (i >= NPTS) return;
    float px = pts[3 * i], py = pts[3 * i + 1], pz = pts[3 * i + 2];
    float tx = T[0]*px + T[1]*py + T[2]*pz  + T[3];
    float ty = T[4]*px + T[5]*py + T[6]*pz  + T[7];
    float tz = T[8]*px + T[9]*py + T[10]*pz + T[11];
    float4 o;
    o.x = tx; o.y = ty; o.z = tz;
    o.w = tx*tx + ty*ty + tz*tz + 1.0f;
    dst[i] = o;
}

// ---------------------------------------------------------------------------
// Directed chamfer on preprocessed float4 clouds. One wave per 16-row strip;
// WMMA K=4 tile per 16 columns; running min kept on SQUARED distances.
// partial[blockIdx.x] = sum of the 16 row-minimum distances (after sqrt).
// ---------------------------------------------------------------------------
__global__ __launch_bounds__(32)
void pe_chamfer_dir(const float4* __restrict__ A4, const float4* __restrict__ B4,
                    float* __restrict__ partial)
{
    const int lane = threadIdx.x;              // full wave: EXEC all 1s for WMMA
    const int half = (lane >= 16) ? 1 : 0;
    const int l16  = lane & 15;
    const int rowBase = blockIdx.x * TILE;

    // A fragment (16x4 f32): lanes 0-15 hold (K0,K1)=(x,y) of row l16;
    // lanes 16-31 hold (K2,K3)=(z,1).
    float4 pa = A4[rowBase + l16];
    v2f afrag;
    afrag.x = half ? pa.z : pa.x;
    afrag.y = half ? 1.0f : pa.y;

    // Row norms for this lane's D elements: D VGPR r -> row M = r + half*8.
    float xx[8];
#pragma unroll
    for (int r = 0; r < 8; r++) xx[r] = A4[rowBase + half * 8 + r].w;

    float mn[8];
#pragma unroll
    for (int r = 0; r < 8; r++) mn[r] = 3.0e38f;

    for (int col = 0; col < NPTS; col += TILE) {
        // B fragment (4x16 f32): same half-split over K rows, column N = l16.
        float4 qb = B4[col + l16];
        v2f bfrag;
        bfrag.x = half ? qb.z : qb.x;
        bfrag.y = half ? 1.0f : qb.y;

        v8f acc = {};
        acc = __builtin_amdgcn_wmma_f32_16x16x4_f32(
            /*neg_a=*/false, afrag, /*neg_b=*/false, bfrag,
            /*c_mod=*/(short)0, acc, /*reuse_a=*/false, /*reuse_b=*/false);

        const float yy = qb.w;
#pragma unroll
        for (int r = 0; r < 8; r++) {
            // squared distance; (+1)+(+1)-2(dot+1) cancels homogeneous terms
            float d2 = (xx[r] + yy) - 2.0f * acc[r];
            mn[r] = fminf(mn[r], d2);
        }
    }

    // Min across the 16 columns held by this half-wave (fixed dataflow -> det.)
#pragma unroll
    for (int m = 1; m <= 8; m <<= 1) {
#pragma unroll
        for (int r = 0; r < 8; r++) mn[r] = fminf(mn[r], __shfl_xor(mn[r], m, 32));
    }
    // sqrt hoisted: min(sqrt(x)) == sqrt(min(x)) (monotone) — 8 sqrts per wave
    float s = 0.f;
#pragma unroll
    for (int r = 0; r < 8; r++) s += sqrtf(fmaxf(mn[r], 0.0f));
    s += __shfl_xor(s, 16, 32);                // rows 0-7 + rows 8-15
    if (lane == 0) partial[blockIdx.x] = s;
}

// ---------------------------------------------------------------------------
// Final reduction (single wave, fixed order -> deterministic).
// ---------------------------------------------------------------------------
__global__ __launch_bounds__(32)
void pe_finalize(const float* __restrict__ pw, const float* __restrict__ partial,
                 float* __restrict__ out)
{
    const int lane = threadIdx.x;
    float sums[4];
#pragma unroll
    for (int d = 0; d < 4; d++) {
        float s = 0.f;
        for (int i = lane; i < NBLK; i += 32) s += partial[d * NBLK + i];
#pragma unroll
        for (int m = 16; m >= 1; m >>= 1) s += __shfl_xor(s, m, 32);
        sums[d] = s;
    }
    if (lane == 0) {
        const float invN = 1.0f / (float)NPTS;
        float base_obj  = (sums[0] + sums[1]) * invN;
        float child_obj = (sums[2] + sums[3]) * invN;
        float all = (pw[0] * base_obj + pw[1] * child_obj) * 0.5f;
        out[0] = all;
        out[1] = base_obj;
        out[2] = child_obj;
    }
}

// ---------------------------------------------------------------------------
extern "C" void kernel_launch(void* const* d_in, const int* in_sizes, int n_in,
                              void* d_out, int out_size, void* d_ws, size_t ws_size,
                              hipStream_t stream)
{
    (void)in_sizes; (void)n_in; (void)out_size; (void)ws_size;

    const float* cam  = (const float*)d_in[0];  // [2,8192,3]
    const float* cad  = (const float*)d_in[1];  // [2,8192,3]
    const float* xyz  = (const float*)d_in[2];  // [1,3] anchor
    const float* rpy  = (const float*)d_in[3];  // [1,3] axis
    const float* pw   = (const float*)d_in[4];  // [2]
    const float* quat = (const float*)d_in[5];  // [4]
    const float* bt   = (const float*)d_in[6];  // [3,1]
    const float* js   = (const float*)d_in[7];  // [1]
    float* out = (float*)d_out;                 // 35 floats
    float* ws  = (float*)d_ws;

    float* Tb   = ws;               // base transform
    float* Tc   = ws + 16;          // rel @ base
    float* Tid  = ws + 32;          // identity
    float* part = ws + WS_PART;     // [4][NBLK]

    float4* cad0p = (float4*)(ws + WS_PTS);                 // cad0 @ Tb
    float4* cam0p = (float4*)(ws + WS_PTS + 1 * WS_CLOUD);  // cam0 (id)
    float4* cad1p = (float4*)(ws + WS_PTS + 2 * WS_CLOUD);  // cad1 @ Tc
    float4* cam1p = (float4*)(ws + WS_PTS + 3 * WS_CLOUD);  // cam1 (id)

    pe_prep<<<1, 1, 0, stream>>>(xyz, rpy, quat, bt, js, ws, out);

    const float* cad1 = cad + (size_t)NPTS * 3;
    const float* cam1 = cam + (size_t)NPTS * 3;

    const int xg = (NPTS + 255) / 256;
    pe_xform<<<xg, 256, 0, stream>>>(cad,  Tb,  cad0p);
    pe_xform<<<xg, 256, 0, stream>>>(cam,  Tid, cam0p);
    pe_xform<<<xg, 256, 0, stream>>>(cad1, Tc,  cad1p);
    pe_xform<<<xg, 256, 0, stream>>>(cam1, Tid, cam1p);

    // part 0 chamfer, both directions
    pe_chamfer_dir<<<NBLK, 32, 0, stream>>>(cad0p, cam0p, part + 0 * NBLK);
    pe_chamfer_dir<<<NBLK, 32, 0, stream>>>(cam0p, cad0p, part + 1 * NBLK);
    // part 1 chamfer, both directions
    pe_chamfer_dir<<<NBLK, 32, 0, stream>>>(cad1p, cam1p, part + 2 * NBLK);
    pe_chamfer_dir<<<NBLK, 32, 0, stream>>>(cam1p, cad1p, part + 3 * NBLK);

    pe_finalize<<<1, 32, 0, stream>>>(pw, part, out);
}